// PatchesKernel_51161650430461
// MI455X (gfx1250) — compile-verified
//
#include <hip/hip_runtime.h>
#include <stdint.h>

// Problem constants (from reference: x = (32,1,512,512) f32, K=8, S=4, P=0)
#define B_   32
#define H_   512
#define W_   512
#define KK   8
#define SS   4
#define OH   127
#define OW   127
#define W4   (W_ / 4)        // 128 float4 per input row
#define ROW4 130             // padded LDS row stride in float4 (2080 B -> 8-bank rotate/row)
#define TILE_F4 (KK * W4)    // 1024 float4 staged per block (16 KB payload)
#define NT   256             // threads per block
#define OUT4_PER_BLK (OW * 16)  // 2032 float4 stores per block

// Native clang vector type (required by __builtin_nontemporal_store; same
// 16-byte layout as float4, lowers to ds_load_b128 / global_store_b128).
typedef __attribute__((ext_vector_type(4))) float v4f;

__global__ __launch_bounds__(NT)
void PatchesKernel_im2col_async(const float* __restrict__ x,
                                float* __restrict__ out) {
    // Padded staging tile: 8 rows x 130 float4 = 16,640 bytes of LDS
    __shared__ v4f tile[KK * ROW4];

    const int t   = threadIdx.x;
    const int blk = blockIdx.x;        // blk = b * OH + oy
    const int b   = blk / OH;
    const int oy  = blk - b * OH;

    // ---------------- Stage 1: async global -> LDS copy of 8 input rows ---------------
    // Band base: x[b][0][oy*4][0]  (byte address, uniform -> SGPR pair)
    const uint64_t src_base =
        (uint64_t)(uintptr_t)x +
        ((uint64_t)(b * H_ + oy * SS) * (uint64_t)W_) * 4ull;

    // LDS byte offset of `tile` (low 32 bits of the shared-aperture address)
    const uint32_t lds_base = (uint32_t)(uintptr_t)(void*)&tile[0];

    #pragma unroll
    for (int i = 0; i < TILE_F4 / NT; ++i) {     // 4 x b128 per thread
        const int j   = i * NT + t;              // tile float4 index 0..1023
        const int row = j >> 7;                  // kernel row 0..7
        const int col = j & 127;                 // float4 column 0..127
        const uint32_t voff = (uint32_t)((row * W4 + col) * 16);          // global byte off
        const uint32_t lds  = lds_base + (uint32_t)((row * ROW4 + col) * 16); // LDS byte addr
        asm volatile("global_load_async_to_lds_b128 %0, %1, %2"
                     :: "v"(lds), "v"(voff), "s"(src_base)
                     : "memory");
    }

#if __has_builtin(__builtin_amdgcn_s_wait_asynccnt)
    __builtin_amdgcn_s_wait_asynccnt(0);
#else
    asm volatile("s_wait_asynccnt 0x0" ::: "memory");
#endif
    __syncthreads();

    // ---------------- Stage 2: coalesced b128 streaming stores ----------------
    // out4 flat index = (b*OH*OW + oy*OW + ox)*16 + c4, with c4 = kr*2 + half and
    // out4[...] == in4 float4 at tile[kr][ox + half]  (exact float4 aliasing: S=4, K=8)
    v4f* __restrict__ out4 = (v4f*)out;
    const long long out_base4 = ((long long)b * (OH * OW) + (long long)oy * OW) * 16;

    for (int idx = t; idx < OUT4_PER_BLK; idx += NT) {   // 8 iters (last partial)
        const int ox   = idx >> 4;
        const int c4   = idx & 15;
        const int kr   = c4 >> 1;
        const int half = c4 & 1;
        const v4f v = tile[kr * ROW4 + ox + half];
        __builtin_nontemporal_store(v, &out4[out_base4 + idx]);
    }
}

extern "C" void kernel_launch(void* const* d_in, const int* in_sizes, int n_in,
                              void* d_out, int out_size, void* d_ws, size_t ws_size,
                              hipStream_t stream) {
    (void)in_sizes; (void)n_in; (void)d_ws; (void)ws_size; (void)out_size;
    const float* x = (const float*)d_in[0];
    float* out = (float*)d_out;

    const int blocks = B_ * OH;   // 4064 blocks, one per (batch, output row)
    PatchesKernel_im2col_async<<<dim3(blocks), dim3(NT), 0, stream>>>(x, out);
}